// MolecularGCNModel_1486058684511
// MI455X (gfx1250) — compile-verified
//
#include <hip/hip_runtime.h>
#include <hip/hip_bf16.h>

#define N_NODES 100000
#define HID 64
#define N_GRAPHS 1024

typedef __attribute__((ext_vector_type(2))) float v2f;
typedef __attribute__((ext_vector_type(8))) float v8f;

// ---------------------------------------------------------------- utilities
__global__ void zero_f32(float* __restrict__ p, long long n) {
  long long i = (long long)blockIdx.x * blockDim.x + threadIdx.x;
  if (i < n) p[i] = 0.0f;
}

__global__ void deg_init(float* __restrict__ deg, int n) {
  int i = blockIdx.x * blockDim.x + threadIdx.x;
  if (i < n) deg[i] = 1.0f;  // self-loop contributes 1 to every node
}

__global__ void deg_accum(const int* __restrict__ col, float* __restrict__ deg, int e) {
  int i = blockIdx.x * blockDim.x + threadIdx.x;
  if (i < e) atomicAdd(&deg[col[i]], 1.0f);
}

__global__ void deg_to_dinv(float* __restrict__ deg, int n) {
  int i = blockIdx.x * blockDim.x + threadIdx.x;
  if (i < n) deg[i] = rsqrtf(deg[i]);  // deg >= 1 always (self loops)
}

// ---------------------------------------------------------------- WMMA GEMM
// Out[M,64] = A[M,K] @ W[K,64], fp32 WMMA 16x16x4. M % 16 == 0, K % 4 == 0.
// One 16x16 output tile per wave; W staged in LDS (max 128*64*4 = 32 KB).
__global__ __launch_bounds__(256) void gemm_wmma_f32(
    const float* __restrict__ A, const float* __restrict__ W,
    float* __restrict__ Out, int M, int K)
{
  __shared__ float sW[128 * 64];
  for (int i = threadIdx.x; i < K * 64; i += 256) sW[i] = W[i];
  __syncthreads();

  const int wave = threadIdx.x >> 5;
  const int lane = threadIdx.x & 31;
  const int ml   = lane & 15;          // row/col within half-wave
  const int kh   = (lane >> 4) * 2;    // K sub-offset: 0 (lanes 0-15) or 2 (lanes 16-31)

  const int tilesM = M >> 4;
  const int total  = tilesM * 4;       // 64/16 = 4 N-tiles

  for (int tile = blockIdx.x * 8 + wave; tile < total; tile += gridDim.x * 8) {
    const int mt = tile >> 2;
    const int nt = tile & 3;
    const int m0 = mt * 16, n0 = nt * 16;

    v8f acc = {};
    const float* arow = A + (size_t)(m0 + ml) * K + kh;
    for (int k = 0; k < K; k += 4) {
      v2f a, b;
      a.x = arow[k + 0];
      a.y = arow[k + 1];
      b.x = sW[(k + kh) * 64 + n0 + ml];
      b.y = sW[(k + kh + 1) * 64 + n0 + ml];
      acc = __builtin_amdgcn_wmma_f32_16x16x4_f32(
          false, a, false, b, (short)0, acc, false, false);
    }
    // C/D layout: VGPR j holds row m0+j (lanes 0-15) / m0+8+j (lanes 16-31)
    const int rbase = m0 + (lane >> 4) * 8;
    const int c = n0 + ml;
#pragma unroll
    for (int j = 0; j < 8; ++j)
      Out[(size_t)(rbase + j) * 64 + c] = acc[j];
  }
}

// ---------------------------------------------------------------- scatter-add
// One wave per (edge or self-loop); each lane handles channels lane, lane+32.
__global__ void scatter_edges(const float* __restrict__ m,
                              const int* __restrict__ row,
                              const int* __restrict__ col,
                              const float* __restrict__ dinv,
                              float* __restrict__ agg, int E, int N)
{
  long long w = ((long long)blockIdx.x * blockDim.x + threadIdx.x) >> 5;
  int lane = threadIdx.x & 31;
  long long total = (long long)E + N;
  if (w >= total) return;
  int r, c;
  if (w < E) { r = row[w]; c = col[w]; }
  else       { r = c = (int)(w - E); }
  float nrm = dinv[r] * dinv[c];
  const float* src = m + (size_t)r * 64;
  float* dst = agg + (size_t)c * 64;
  atomicAdd(&dst[lane],      src[lane]      * nrm);
  atomicAdd(&dst[lane + 32], src[lane + 32] * nrm);
}

// ---------------------------------------------------------------- batchnorm
// stats[0..63] = sum per channel, stats[64..127] = sum of squares
__global__ __launch_bounds__(256) void bn_stats(const float* __restrict__ h,
                                                float* __restrict__ stats, int N)
{
  const int ch = threadIdx.x & 63;  // channel stays fixed: stride % 64 == 0
  float s = 0.0f, sq = 0.0f;
  long long total = (long long)N * 64;
  for (long long idx = (long long)blockIdx.x * blockDim.x + threadIdx.x;
       idx < total; idx += (long long)gridDim.x * blockDim.x) {
    float v = h[idx];
    s += v; sq += v * v;
  }
  __shared__ float ls[64], lq[64];
  if (threadIdx.x < 64) { ls[threadIdx.x] = 0.0f; lq[threadIdx.x] = 0.0f; }
  __syncthreads();
  atomicAdd(&ls[ch], s);
  atomicAdd(&lq[ch], sq);
  __syncthreads();
  if (threadIdx.x < 64) {
    atomicAdd(&stats[threadIdx.x],      ls[threadIdx.x]);
    atomicAdd(&stats[64 + threadIdx.x], lq[threadIdx.x]);
  }
}

// h = relu((h - mean) * rsqrt(var+eps) * gamma + beta [+ res]), in place.
// GCN bias b cancels exactly inside training-mode BN, so it is omitted.
__global__ void bn_apply_relu(float* __restrict__ h,
                              const float* __restrict__ stats,
                              const float* __restrict__ gamma,
                              const float* __restrict__ beta,
                              const float* __restrict__ res, int N)
{
  long long idx = (long long)blockIdx.x * blockDim.x + threadIdx.x;
  long long total = (long long)N * 64;
  if (idx >= total) return;
  int ch = (int)(idx & 63);
  float invn  = 1.0f / (float)N;
  float mean  = stats[ch] * invn;
  float var   = stats[64 + ch] * invn - mean * mean;
  float scale = gamma[ch] * rsqrtf(var + 1e-5f);
  float shift = beta[ch] - mean * scale;
  float v = h[idx] * scale + shift;
  if (res) v += res[idx];
  h[idx] = fmaxf(v, 0.0f);
}

// ---------------------------------------------------------------- pooling
__global__ void pool_accum(const float* __restrict__ h,
                           const int* __restrict__ batch,
                           float* __restrict__ pool,
                           float* __restrict__ cnt, int N)
{
  long long node = ((long long)blockIdx.x * blockDim.x + threadIdx.x) >> 5;
  int lane = threadIdx.x & 31;
  if (node >= N) return;
  int g = batch[node];
  const float* src = h + (size_t)node * 64;
  float* dst = pool + (size_t)g * 64;
  atomicAdd(&dst[lane],      src[lane]);
  atomicAdd(&dst[lane + 32], src[lane + 32]);
  if (lane == 0) atomicAdd(&cnt[g], 1.0f);
}

__global__ void pool_final(const float* __restrict__ pool,
                           const float* __restrict__ cnt,
                           const float* __restrict__ fc_w,
                           const float* __restrict__ fc_b,
                           float* __restrict__ out, int G)
{
  int g = (blockIdx.x * blockDim.x + threadIdx.x) >> 5;
  int lane = threadIdx.x & 31;
  if (g >= G) return;
  float inv = 1.0f / fmaxf(cnt[g], 1.0f);
  const float* p = pool + (size_t)g * 64;
  float partial = p[lane] * fc_w[lane] + p[lane + 32] * fc_w[lane + 32];
  partial *= inv;
  for (int off = 16; off > 0; off >>= 1)
    partial += __shfl_down(partial, off, 32);
  if (lane == 0) out[g] = partial + fc_b[0];
}

// ---------------------------------------------------------------- driver
extern "C" void kernel_launch(void* const* d_in, const int* in_sizes, int n_in,
                              void* d_out, int out_size, void* d_ws, size_t ws_size,
                              hipStream_t stream) {
  const float* x     = (const float*)d_in[0];
  const int*   ei    = (const int*)d_in[1];
  const int*   batch = (const int*)d_in[2];
  const float* w1    = (const float*)d_in[3];
  const float* g1    = (const float*)d_in[5];
  const float* be1   = (const float*)d_in[6];
  const float* w2    = (const float*)d_in[7];
  const float* g2    = (const float*)d_in[9];
  const float* be2   = (const float*)d_in[10];
  const float* w3    = (const float*)d_in[11];
  const float* g3    = (const float*)d_in[13];
  const float* be3   = (const float*)d_in[14];
  const float* fcw   = (const float*)d_in[15];
  const float* fcb   = (const float*)d_in[16];
  float* out = (float*)d_out;

  const int N = in_sizes[0] / 128;   // 100000
  const int E = in_sizes[1] / 2;     // 1600000
  const int G = out_size;            // 1024
  const int IN_DIM = 128;
  const int* rowp = ei;
  const int* colp = ei + E;

  float* dinv  = (float*)d_ws;                      // N
  float* bufA  = dinv  + (((size_t)N + 63) & ~63ull);  // N*64 (messages m)
  float* bufB  = bufA  + (size_t)N * 64;            // N*64
  float* bufC  = bufB  + (size_t)N * 64;            // N*64
  float* stats = bufC  + (size_t)N * 64;            // 128
  float* pool  = stats + 128;                       // G*64
  float* cnt   = pool  + (size_t)G * 64;            // G

  const long long NH = (long long)N * 64;
  const int TB = 256;
  dim3 blk(TB);
  unsigned gN    = (N + TB - 1) / TB;
  unsigned gE    = (E + TB - 1) / TB;
  unsigned gNH   = (unsigned)((NH + TB - 1) / TB);
  unsigned gScat = (unsigned)((((long long)E + N) * 32 + TB - 1) / TB);
  unsigned gPool = (unsigned)(((long long)N * 32 + TB - 1) / TB);
  unsigned gGemm = 2048;

  // degree -> dinv
  deg_init<<<gN, blk, 0, stream>>>(dinv, N);
  deg_accum<<<gE, blk, 0, stream>>>(colp, dinv, E);
  deg_to_dinv<<<gN, blk, 0, stream>>>(dinv, N);

  // ---- layer 1: bufA = x@w1 ; bufB = agg ; BN+ReLU in place -> h1 (residual)
  gemm_wmma_f32<<<gGemm, blk, 0, stream>>>(x, w1, bufA, N, IN_DIM);
  zero_f32<<<gNH, blk, 0, stream>>>(bufB, NH);
  scatter_edges<<<gScat, blk, 0, stream>>>(bufA, rowp, colp, dinv, bufB, E, N);
  zero_f32<<<1, 128, 0, stream>>>(stats, 128);
  bn_stats<<<512, blk, 0, stream>>>(bufB, stats, N);
  bn_apply_relu<<<gNH, blk, 0, stream>>>(bufB, stats, g1, be1, nullptr, N);

  // ---- layer 2: bufA = h1@w2 ; bufC = agg ; BN + residual(h1) + ReLU -> h2
  gemm_wmma_f32<<<gGemm, blk, 0, stream>>>(bufB, w2, bufA, N, HID);
  zero_f32<<<gNH, blk, 0, stream>>>(bufC, NH);
  scatter_edges<<<gScat, blk, 0, stream>>>(bufA, rowp, colp, dinv, bufC, E, N);
  zero_f32<<<1, 128, 0, stream>>>(stats, 128);
  bn_stats<<<512, blk, 0, stream>>>(bufC, stats, N);
  bn_apply_relu<<<gNH, blk, 0, stream>>>(bufC, stats, g2, be2, bufB, N);

  // ---- layer 3: bufA = h2@w3 ; bufB = agg (h1 dead) ; BN+ReLU -> h3
  gemm_wmma_f32<<<gGemm, blk, 0, stream>>>(bufC, w3, bufA, N, HID);
  zero_f32<<<gNH, blk, 0, stream>>>(bufB, NH);
  scatter_edges<<<gScat, blk, 0, stream>>>(bufA, rowp, colp, dinv, bufB, E, N);
  zero_f32<<<1, 128, 0, stream>>>(stats, 128);
  bn_stats<<<512, blk, 0, stream>>>(bufB, stats, N);
  bn_apply_relu<<<gNH, blk, 0, stream>>>(bufB, stats, g3, be3, nullptr, N);

  // ---- global mean pool + fc
  zero_f32<<<(unsigned)(((long long)G * 65 + TB - 1) / TB), blk, 0, stream>>>(pool, (long long)G * 65);
  pool_accum<<<gPool, blk, 0, stream>>>(bufB, batch, pool, cnt, N);
  pool_final<<<(G * 32 + TB - 1) / TB, blk, 0, stream>>>(pool, cnt, fcw, fcb, out, G);
}